// Attention_51814485459027
// MI455X (gfx1250) — compile-verified
//
#include <hip/hip_runtime.h>

#define B_ 32
#define S_ 4096
#define H_ 512
#define NCHUNK 16
#define ROWS_PER_CHUNK (S_ / NCHUNK)                       // 256
#define WAVES_PER_BLOCK 8
#define ROWS_PER_WAVE (ROWS_PER_CHUNK / WAVES_PER_BLOCK)   // 32

typedef __attribute__((ext_vector_type(2))) float v2f;
typedef __attribute__((ext_vector_type(8))) float v8f;

// ---------------------------------------------------------------------------
// Kernel 1: query[b,o] = sum_h dec[b,h] * W[o,h]   (32x512, K=512), fp32 WMMA.
// One wave per 16x16 output tile. V_WMMA_F32_16X16X4_F32:
//   A (16x4):  lane l, elem i -> M = l&15, K = kbase + 2*(l>>4) + i
//   B (4x16):  lane l, elem i -> N = l&15, K = kbase + 2*(l>>4) + i
//   D (16x16): VGPR v, lane l -> M = v + 8*(l>>4), N = l&15
// ---------------------------------------------------------------------------
__global__ __launch_bounds__(32) void query_wmma_kernel(
    const float* __restrict__ dec, const float* __restrict__ W,
    float* __restrict__ q) {
  const int m0   = blockIdx.x * 16;   // batch tile base (0 or 16)
  const int n0   = blockIdx.y * 16;   // output-feature tile base
  const int lane = threadIdx.x;       // 0..31 (full wave, EXEC all 1s)
  const int half = lane >> 4;         // 0/1
  const int r    = lane & 15;
  const int koff = half * 2;

  v8f c = {};
  for (int k = 0; k < H_; k += 4) {
    const float* ap = dec + (m0 + r) * H_ + k + koff;   // A[m,k] = dec[b,h]
    const float* bp = W   + (n0 + r) * H_ + k + koff;   // B[k,n] = W[o,h]
    v2f a, b;
    a.x = ap[0]; a.y = ap[1];
    b.x = bp[0]; b.y = bp[1];
    c = __builtin_amdgcn_wmma_f32_16x16x4_f32(
        /*neg_a=*/false, a, /*neg_b=*/false, b,
        /*c_mod=*/(short)0, c, /*reuse_a=*/false, /*reuse_b=*/false);
  }

  #pragma unroll
  for (int v = 0; v < 8; ++v) {
    const int m = v + 8 * half;                   // batch index within tile
    q[(m0 + m) * H_ + (n0 + r)] = c[v];
  }
}

// ---------------------------------------------------------------------------
// Kernel 2: fused scores+context, single streaming pass over enc (HBM-bound).
// Per wave: 32 rows. Lane l owns h-positions {128*j + 4*l + c}, j=0..3, c=0..3
// for q, the streamed enc row, and the ctx accumulator (same layout, so the
// row stays in registers between the dot and the rank-1 update).
// ---------------------------------------------------------------------------
__global__ __launch_bounds__(256) void fused_scores_context(
    const float* __restrict__ enc, const float* __restrict__ qws,
    float* __restrict__ partials) {
  const int b     = blockIdx.x;       // 0..31
  const int chunk = blockIdx.y;       // 0..15
  const int tid   = threadIdx.x;      // 0..255
  const int wave  = tid >> 5;         // 0..7
  const int lane  = tid & 31;

  const float* q = qws + b * H_;
  float4 qr[4];
  #pragma unroll
  for (int j = 0; j < 4; ++j)
    qr[j] = *(const float4*)(q + j * 128 + lane * 4);

  float4 ctx[4];
  #pragma unroll
  for (int j = 0; j < 4; ++j) ctx[j] = make_float4(0.f, 0.f, 0.f, 0.f);

  const int s0 = chunk * ROWS_PER_CHUNK + wave * ROWS_PER_WAVE;
  const float* base = enc + ((size_t)b * S_ + s0) * H_;

  for (int s = 0; s < ROWS_PER_WAVE; ++s) {
    const float* row = base + (size_t)s * H_;
    // Prefetch next row's slice (gfx1250 global_prefetch_b8).
    __builtin_prefetch(row + H_ + lane * 4, 0, 0);

    float4 e[4];
    #pragma unroll
    for (int j = 0; j < 4; ++j)
      e[j] = *(const float4*)(row + j * 128 + lane * 4);

    float p = 0.f;
    #pragma unroll
    for (int j = 0; j < 4; ++j)
      p += e[j].x * qr[j].x + e[j].y * qr[j].y
         + e[j].z * qr[j].z + e[j].w * qr[j].w;

    // wave32 butterfly reduction -> every lane holds the full dot (score).
    #pragma unroll
    for (int off = 16; off >= 1; off >>= 1)
      p += __shfl_xor(p, off, 32);

    #pragma unroll
    for (int j = 0; j < 4; ++j) {
      ctx[j].x += p * e[j].x;
      ctx[j].y += p * e[j].y;
      ctx[j].z += p * e[j].z;
      ctx[j].w += p * e[j].w;
    }
  }

  // Deterministic block reduction across the 8 waves via LDS (16 KB).
  __shared__ float red[WAVES_PER_BLOCK][H_];
  #pragma unroll
  for (int j = 0; j < 4; ++j)
    *(float4*)&red[wave][j * 128 + lane * 4] = ctx[j];
  __syncthreads();

  for (int h = tid; h < H_; h += 256) {
    float sum = 0.f;
    #pragma unroll
    for (int w = 0; w < WAVES_PER_BLOCK; ++w) sum += red[w][h];
    partials[(size_t)(b * NCHUNK + chunk) * H_ + h] = sum;
  }
}

// ---------------------------------------------------------------------------
// Kernel 3: fixed-order reduction of the 16 chunk partials -> context[B,1,H].
// ---------------------------------------------------------------------------
__global__ __launch_bounds__(256) void reduce_context(
    const float* __restrict__ partials, float* __restrict__ out) {
  const int b = blockIdx.x;
  for (int h = threadIdx.x; h < H_; h += 256) {
    float sum = 0.f;
    #pragma unroll
    for (int c = 0; c < NCHUNK; ++c)
      sum += partials[(size_t)(b * NCHUNK + c) * H_ + h];
    out[b * H_ + h] = sum;
  }
}

extern "C" void kernel_launch(void* const* d_in, const int* in_sizes, int n_in,
                              void* d_out, int out_size, void* d_ws, size_t ws_size,
                              hipStream_t stream) {
  (void)in_sizes; (void)n_in; (void)out_size; (void)ws_size;
  const float* enc = (const float*)d_in[0];   // [B,S,H]
  const float* dec = (const float*)d_in[1];   // [B,1,H]
  const float* W   = (const float*)d_in[2];   // [H,H]
  float* out = (float*)d_out;                 // [B,1,H]

  float* q_ws     = (float*)d_ws;             // B*H floats       (64 KB)
  float* partials = q_ws + B_ * H_;           // B*NCHUNK*H floats (1 MB)

  query_wmma_kernel<<<dim3(2, H_ / 16), 32, 0, stream>>>(dec, W, q_ws);
  fused_scores_context<<<dim3(B_, NCHUNK), 256, 0, stream>>>(enc, q_ws, partials);
  reduce_context<<<B_, 256, 0, stream>>>(partials, out);
}